// MultiViewCausalContrastiveLoss_38285338477179
// MI455X (gfx1250) — compile-verified
//
#include <hip/hip_runtime.h>
#include <hip/hip_bf16.h>

typedef __attribute__((ext_vector_type(2))) float v2f;
typedef __attribute__((ext_vector_type(8))) float v8f;

#define MARGIN   0.4f
#define NEG_INF  -1e9f
#define NORM_EPS 1e-12f

// Each wave handles 2 batch elements: rows 0-5 of the 16-row WMMA tile are
// b0's {opt0..3, evt, evi}, rows 8-13 are b1's. Gram = A * A^T accumulated
// with V_WMMA_F32_16X16X4_F32 over K = H in chunks of 4.
__global__ __launch_bounds__(256) void mvccl_main(
    const float* __restrict__ opt,
    const float* __restrict__ evt,
    const float* __restrict__ evi,
    const long long* __restrict__ labels,
    float* __restrict__ partial,   // 2 floats per block: (total, count)
    int B, int H)
{
    const int tid   = threadIdx.x;
    const int lane  = tid & 31;
    const int wave  = tid >> 5;
    const int bBase = blockIdx.x * 16 + wave * 2;   // this wave's first batch idx

    // A-matrix 16x4 f32 layout: lanes 0-15 -> row=lane, K=0,1 ; lanes 16-31 -> row=lane-16, K=2,3
    const int m    = lane & 15;         // tile row 0..15
    const int half = m >> 3;            // 0 -> bBase, 1 -> bBase+1
    const int r    = m & 7;             // 0-3: option r, 4: event, 5: evidence, 6-7: zero pad
    const int koff = (lane >> 4) * 2;   // K offset within chunk
    const int b    = bBase + half;
    const bool lvalid = (r < 6) && (b < B);

    const float* base = opt;            // safe dummy for padded/out-of-range lanes
    if (lvalid) {
        if (r < 4)       base = opt + ((size_t)b * 4 + r) * (size_t)H;
        else if (r == 4) base = evt + (size_t)b * (size_t)H;
        else             base = evi + (size_t)b * (size_t)H;
    }
    base += koff;
    const float mask = lvalid ? 1.0f : 0.0f;

    v8f acc = {};
    #pragma unroll 4
    for (int k = 0; k + 3 < H; k += 4) {
        const v2f d = *reinterpret_cast<const v2f*>(base + k);  // 8B, aligned
        v2f a;
        a.x = d.x * mask;   // zero padded rows so they contribute nothing
        a.y = d.y * mask;
        // D = A * B + C with B fragment == A fragment  =>  Gram accumulate
        acc = __builtin_amdgcn_wmma_f32_16x16x4_f32(
            /*neg_a=*/false, a, /*neg_b=*/false, a,
            /*c_mod=*/(short)0, acc, /*reuse_a=*/false, /*reuse_b=*/false);
    }

    // C layout: lanes 0-15: C[v][lane] for v=0..7 ; lanes 16-31: C[v+8][lane-16]
    // b0 Gram G0[m][n] = acc[m] on lane n        (m,n in 0..5)
    // b1 Gram G1[m][n] = acc[m] on lane 24+n     (rows/cols 8..13)
    __shared__ float G[8][2][36];
    __shared__ float sTot[16];
    __shared__ float sCnt[16];

    if (lane < 6) {
        #pragma unroll
        for (int mm = 0; mm < 6; ++mm) G[wave][0][mm * 6 + lane] = acc[mm];
    }
    if (lane >= 24 && lane < 30) {
        #pragma unroll
        for (int mm = 0; mm < 6; ++mm) G[wave][1][mm * 6 + (lane - 24)] = acc[mm];
    }
    __syncthreads();

    if (lane < 2) {
        const int bb = bBase + lane;
        float tot = 0.0f, cnt = 0.0f;
        if (bb < B) {
            const float* g = G[wave][lane];
            float inv[6];
            #pragma unroll
            for (int j = 0; j < 6; ++j)
                inv[j] = 1.0f / fmaxf(sqrtf(g[j * 6 + j]), NORM_EPS);

            const long long* lb = labels + (size_t)bb * 4;
            bool pos[4], neg[4];
            int npos = 0, nneg = 0;
            #pragma unroll
            for (int k = 0; k < 4; ++k) {
                const long long lv = lb[k];
                pos[k] = (lv == 1);
                neg[k] = (lv == 0);
                if (pos[k]) npos++;
                if (neg[k]) nneg++;
            }
            const float nposf  = (float)npos;
            const bool  vb     = (npos > 0) && (nneg > 0);
            const float validf = vb ? 1.0f : 0.0f;

            // Views 1 & 2: options vs event (row 4) / evidence (row 5)
            float tv = 0.0f;
            #pragma unroll
            for (int ref = 4; ref <= 5; ++ref) {
                float pm = 0.0f, hn = NEG_INF;
                #pragma unroll
                for (int k = 0; k < 4; ++k) {
                    const float s = g[k * 6 + ref] * inv[k] * inv[ref];
                    if (pos[k]) pm += s;
                    if (neg[k]) hn = fmaxf(hn, s);
                }
                pm /= fmaxf(nposf, 1.0f);
                tv += fmaxf(hn - pm + MARGIN, 0.0f);
            }

            // View 3: cross-option (neg mask includes q==p, matching reference)
            float v3 = 0.0f;
            #pragma unroll
            for (int p = 0; p < 4; ++p) {
                float ps = 0.0f, hn = NEG_INF;
                #pragma unroll
                for (int q = 0; q < 4; ++q) {
                    const float s = g[p * 6 + q] * inv[p] * inv[q];
                    if (q != p && pos[q]) ps += s;
                    if (neg[q]) hn = fmaxf(hn, s);
                }
                const float psim = (npos > 1) ? ps / fmaxf(nposf - 1.0f, 1.0f) : 1.0f;
                const float h = fmaxf(hn - psim + MARGIN, 0.0f);
                if (pos[p]) v3 += h;
            }

            tot = (tv + v3) * validf;
            cnt = validf * (2.0f + nposf);
        }
        sTot[wave * 2 + lane] = tot;
        sCnt[wave * 2 + lane] = cnt;
    }
    __syncthreads();

    if (tid == 0) {
        float t = 0.0f, c = 0.0f;
        #pragma unroll
        for (int i = 0; i < 16; ++i) { t += sTot[i]; c += sCnt[i]; }
        partial[(size_t)blockIdx.x * 2 + 0] = t;
        partial[(size_t)blockIdx.x * 2 + 1] = c;
    }
}

// Deterministic serial final reduction (every partial slot is written by
// mvccl_main, so no workspace zero-init is needed).
__global__ void mvccl_finish(const float* __restrict__ partial,
                             float* __restrict__ out, int nblocks)
{
    if (threadIdx.x == 0 && blockIdx.x == 0) {
        float t = 0.0f, c = 0.0f;
        for (int i = 0; i < nblocks; ++i) {
            t += partial[2 * i + 0];
            c += partial[2 * i + 1];
        }
        out[0] = t / fmaxf(c, 1.0f);
    }
}

extern "C" void kernel_launch(void* const* d_in, const int* in_sizes, int n_in,
                              void* d_out, int out_size, void* d_ws, size_t ws_size,
                              hipStream_t stream)
{
    const float*     opt    = (const float*)d_in[0];      // (B,4,H) f32
    const float*     evt    = (const float*)d_in[1];      // (B,H)   f32
    const float*     evi    = (const float*)d_in[2];      // (B,H)   f32
    const long long* labels = (const long long*)d_in[3];  // (B,4)   i64

    const int B = in_sizes[3] / 4;
    const int H = in_sizes[0] / in_sizes[3];              // = 1024

    const int nblocks = (B + 15) / 16;                    // 16 batch per 256-thread block
    float* partial = (float*)d_ws;                        // 2*nblocks floats

    mvccl_main<<<nblocks, 256, 0, stream>>>(opt, evt, evi, labels, partial, B, H);
    mvccl_finish<<<1, 32, 0, stream>>>(partial, (float*)d_out, nblocks);
}